// ReconstructionUpdateLayer_28802050687292
// MI455X (gfx1250) — compile-verified
//
#include <hip/hip_runtime.h>
#include <stdint.h>

// ---------------------------------------------------------------------------
// ReconstructionUpdateLayer for MI455X (gfx1250, wave32)
//
//   spc  = FFT_h( IFFT_w( nnline_input ) )            [B,1024,1024] c64
//   tmp  = sum_h conj(spc)*input   -> phi = -atan2(re,im)
//   rec  = FFT_w( IFFT_h( input * e^{-i phi} , n=2048 ), n=2048 )  zero-padded
//   out  = minus*(rho/(1+rho)) + rec*(1/(1+rho)),  minus = nnl_out - mul_out
//
// FFTs: per-block LDS four-step N = N2*16; inner radix-2 Stockham (VALU),
// twiddle, outer radix-16 as complex 16x16 matmul on V_WMMA_F32_16X16X4_F32.
// Workspace requirement: 32MiB (spc) + 64KiB (scale) + 64MiB (rec1) ~= 96MiB.
// ---------------------------------------------------------------------------

#define BATCH 4
#define DH    1024
#define DW    1024
#define FRH   2048
#define FRW   2048

static constexpr float kC1  = 0.1f / 1.1f;   // rho/(1+rho)
static constexpr float kC2  = 1.0f / 1.1f;   // 1/(1+rho)
static constexpr float PI_F = 3.14159265358979323846f;

struct c32 { float x, y; };
__device__ __forceinline__ c32 cmul(c32 a, c32 b){ return {a.x*b.x - a.y*b.y, a.x*b.y + a.y*b.x}; }
__device__ __forceinline__ c32 cadd(c32 a, c32 b){ return {a.x+b.x, a.y+b.y}; }
__device__ __forceinline__ c32 csub(c32 a, c32 b){ return {a.x-b.x, a.y-b.y}; }

typedef float v2f __attribute__((ext_vector_type(2)));
typedef float v8f __attribute__((ext_vector_type(8)));

#if defined(__gfx1250__) && __has_builtin(__builtin_amdgcn_wmma_f32_16x16x4_f32)
#define USE_WMMA_F32 1
#else
#define USE_WMMA_F32 0
#endif

// ---------------------------------------------------------------------------
// Block-cooperative N-point complex FFT in LDS (256 threads = 8 waves).
// Decomposition: n = n1 + 16*n2, k = k2 + N2*k1, N2 = N/16.
//   inner:  for each n1 in [0,16): N2-pt Stockham DFT of x[n1::16] (stride-16
//           view over natural-order storage -> no permute needed)
//   twiddle: Y[n1][k2] *= exp(sgn*2pi*i*n1*k2/N)
//   outer:  X[k2 + N2*k1] = sum_n1 W16^{n1*k1} Y[n1][k2]
//           == D(16x16) = W16(16x16) x Ytile(16x16) per 16-col tile, one tile
//           per wave, chained V_WMMA_F32_16X16X4_F32 (4 K-slices, 4 real
//           matmuls for complex). Lane layouts per CDNA5 ISA 7.12.2.
// Returns pointer to the buffer holding the ordered result. sgn=-1 forward,
// +1 inverse (caller applies 1/N).
// ---------------------------------------------------------------------------
template<int N, bool INV>
__device__ c32* block_fft(c32* b0, c32* b1)
{
    constexpr int  N2   = N / 16;
    constexpr int  NS2  = N2 / 2;
    constexpr float sgn = INV ? 1.0f : -1.0f;
    constexpr int  NTHR = 256;
    const int tid = threadIdx.x;

    c32* src = b0;
    c32* dst = b1;

    // ---- inner: 16 independent strided Stockham radix-2 FFTs --------------
    int p = 0;
    for (int ns = 1; ns < N2; ns <<= 1, ++p) {
        __syncthreads();
        for (int t = tid; t < N / 2; t += NTHR) {
            int seg = t / NS2;                 // n1
            int tt  = t - seg * NS2;
            int j   = tt & (ns - 1);
            c32 a = src[seg + 16 * tt];
            c32 b = src[seg + 16 * (tt + NS2)];
            float ang = sgn * PI_F * (float)j / (float)ns;
            float sn, cs; __sincosf(ang, &sn, &cs);
            c32 wb = cmul({cs, sn}, b);
            int o = ((tt >> p) << (p + 1)) + j;
            dst[seg + 16 * o]        = cadd(a, wb);
            dst[seg + 16 * (o + ns)] = csub(a, wb);
        }
        c32* t2 = src; src = dst; dst = t2;
    }
    __syncthreads();

    // ---- twiddle: Y[n1][k2] *= W_N^{sgn*n1*k2} ----------------------------
    for (int idx = tid; idx < N; idx += NTHR) {
        int n1 = idx & 15;
        int k2 = idx >> 4;
        if (n1 != 0 && k2 != 0) {
            float ang = sgn * (2.0f * PI_F) * (float)(n1 * k2) / (float)N;
            float sn, cs; __sincosf(ang, &sn, &cs);
            src[idx] = cmul(src[idx], {cs, sn});
        }
    }
    __syncthreads();

    // ---- outer radix-16 via WMMA ------------------------------------------
    const int wave = tid >> 5;
    const int lane = tid & 31;
    constexpr int NTILE = N2 / 16;             // 4 (N=1024) or 8 (N=2048)
#if USE_WMMA_F32
    if (wave < NTILE) {
        const int m   = lane & 15;             // A: M row / B: N col (ISA layout)
        const int khi = (lane < 16) ? 0 : 2;   // K sub-offset within slice
        const int col = wave * 16 + m;         // global k2 of this lane's column
        // A = W16[k1][n1] = exp(sgn*2pi*i*m*k/16), packed in 16x4-f32 A layout.
        v2f aRe[4], aIm[4];
#pragma unroll
        for (int s = 0; s < 4; ++s) {
            float k0 = (float)(4 * s + khi);
            float a0 = sgn * (PI_F / 8.0f) * (float)m * k0;
            float a1 = sgn * (PI_F / 8.0f) * (float)m * (k0 + 1.0f);
            float s0, c0, s1, c1;
            __sincosf(a0, &s0, &c0);
            __sincosf(a1, &s1, &c1);
            aRe[s][0] = c0; aRe[s][1] = c1;
            aIm[s][0] = s0; aIm[s][1] = s1;
        }
        v8f cr = {};
        v8f ci = {};
#pragma unroll
        for (int s = 0; s < 4; ++s) {
            int r0 = 4 * s + khi;              // n1 index of this lane's B elems
            c32 y0 = src[r0     + 16 * col];
            c32 y1 = src[r0 + 1 + 16 * col];
            v2f br = {  y0.x,  y1.x };
            v2f bi = {  y0.y,  y1.y };
            v2f bn = { -y0.y, -y1.y };
            // Dr += Ar*Br - Ai*Bi ; Di += Ar*Bi + Ai*Br
            cr = __builtin_amdgcn_wmma_f32_16x16x4_f32(false, aRe[s], false, br, (short)0, cr, false, false);
            cr = __builtin_amdgcn_wmma_f32_16x16x4_f32(false, aIm[s], false, bn, (short)0, cr, false, false);
            ci = __builtin_amdgcn_wmma_f32_16x16x4_f32(false, aRe[s], false, bi, (short)0, ci, false, false);
            ci = __builtin_amdgcn_wmma_f32_16x16x4_f32(false, aIm[s], false, br, (short)0, ci, false, false);
        }
        // D layout: VGPR r -> M = r + (lane<16?0:8), N = lane&15
#pragma unroll
        for (int r = 0; r < 8; ++r) {
            int k1 = r + ((lane < 16) ? 0 : 8);
            dst[col + N2 * k1] = { cr[r], ci[r] };
        }
    }
#else
    // VALU fallback (also used for host pass / non-gfx1250)
    for (int k = tid; k < N; k += NTHR) {
        int k1 = k / N2;
        int k2 = k - k1 * N2;
        c32 acc = {0.f, 0.f};
#pragma unroll
        for (int n1 = 0; n1 < 16; ++n1) {
            float ang = sgn * (PI_F / 8.0f) * (float)(n1 * k1);
            float sn, cs; __sincosf(ang, &sn, &cs);
            acc = cadd(acc, cmul({cs, sn}, src[n1 + 16 * k2]));
        }
        dst[k] = acc;
    }
#endif
    __syncthreads();
    return dst;
}

// ---------------------------------------------------------------------------
// K1: IFFT along width (contiguous rows) of nnline_input -> spc_rows
// ---------------------------------------------------------------------------
__global__ __launch_bounds__(256)
void k1_row_ifft1024(const float* __restrict__ nnline_input, c32* __restrict__ spc)
{
    __shared__ c32 sA[1024], sB[1024];
    const int line = blockIdx.x;                       // b*DH + h
    const float* row = nnline_input + (size_t)line * DW;
    for (int w = threadIdx.x; w < DW; w += 256) sA[w] = { row[w], 0.f };
    __syncthreads();
    c32* res = block_fft<1024, true>(sA, sB);
    c32* out = spc + (size_t)line * DW;
    constexpr float inv = 1.0f / 1024.0f;
    for (int w = threadIdx.x; w < DW; w += 256) out[w] = { res[w].x * inv, res[w].y * inv };
}

// ---------------------------------------------------------------------------
// K2: FFT along height per column + inner product with input -> e^{-i*phi}
// ---------------------------------------------------------------------------
__global__ __launch_bounds__(256)
void k2_col_fft_phi(const c32* __restrict__ spc, const float* __restrict__ inp,
                    c32* __restrict__ escale)
{
    __shared__ c32 sA[1024], sB[1024];
    const int b = blockIdx.x / DW;
    const int w = blockIdx.x - b * DW;
    const c32* base = spc + (size_t)b * DH * DW + w;
    for (int h = threadIdx.x; h < DH; h += 256) sA[h] = base[(size_t)h * DW];
    __syncthreads();
    c32* res = block_fft<1024, false>(sA, sB);
    c32* red = (res == sA) ? sB : sA;                  // free ping-pong buffer
    const float* icol = inp + (size_t)b * DH * DW + w;
    c32 part = {0.f, 0.f};
    for (int h = threadIdx.x; h < DH; h += 256) {
        float v = icol[(size_t)h * DW];
        part.x += res[h].x * v;                        // Re(conj(spc)*in)
        part.y -= res[h].y * v;                        // Im(conj(spc)*in)
    }
    red[threadIdx.x] = part;
    __syncthreads();
    for (int off = 128; off > 0; off >>= 1) {
        if (threadIdx.x < off) red[threadIdx.x] = cadd(red[threadIdx.x], red[threadIdx.x + off]);
        __syncthreads();
    }
    if (threadIdx.x == 0) {
        c32 t = red[0];
        float phi = -atan2f(t.x, t.y);
        float sn, cs; __sincosf(phi, &sn, &cs);
        escale[(size_t)b * DW + w] = { cs, -sn };      // exp(-i*phi)
    }
}

// ---------------------------------------------------------------------------
// K3: rec0 = input * e^{-i phi}; zero-pad column to 2048; IFFT along height.
// ---------------------------------------------------------------------------
__global__ __launch_bounds__(256)
void k3_col_ifft2048(const float* __restrict__ inp, const c32* __restrict__ escale,
                     c32* __restrict__ rec1)
{
    __shared__ c32 sA[2048], sB[2048];
    const int b = blockIdx.x / DW;
    const int w = blockIdx.x - b * DW;
    const c32 sc = escale[(size_t)b * DW + w];
    const float* icol = inp + (size_t)b * DH * DW + w;
    for (int h = threadIdx.x; h < DH; h += 256) {
        float v = icol[(size_t)h * DW];
        sA[h] = { v * sc.x, v * sc.y };
    }
    for (int h = DH + threadIdx.x; h < FRH; h += 256) sA[h] = {0.f, 0.f};
    __syncthreads();
    c32* res = block_fft<2048, true>(sA, sB);
    c32* ocol = rec1 + (size_t)b * FRH * DW + w;
    constexpr float inv = 1.0f / 2048.0f;
    for (int h = threadIdx.x; h < FRH; h += 256)
        ocol[(size_t)h * DW] = { res[h].x * inv, res[h].y * inv };
}

// ---------------------------------------------------------------------------
// K4: zero-padded FFT along width (contiguous rows, async-to-LDS load) + blend
// ---------------------------------------------------------------------------
__global__ __launch_bounds__(256)
void k4_row_fft2048_combine(const c32* __restrict__ rec1,
                            const float* __restrict__ nnl_out,
                            const float* __restrict__ mul_out,
                            c32* __restrict__ out)
{
    __shared__ c32 sA[2048], sB[2048];
    const int line = blockIdx.x;                       // b*FRH + h
    const c32* row = rec1 + (size_t)line * DW;
#if defined(__gfx1250__)
    // CDNA5 async global->LDS: 16B per lane, 512 x b128 covers the 8KiB row.
    for (int i = threadIdx.x; i < DW / 2; i += 256) {
        unsigned lds = (unsigned)(uintptr_t)(&sA[2 * i]);
        const c32* g = row + 2 * i;
        asm volatile("global_load_async_to_lds_b128 %0, %1, off"
                     :: "v"(lds), "v"(g) : "memory");
    }
    for (int w = DW + threadIdx.x; w < FRW; w += 256) sA[w] = {0.f, 0.f};
    asm volatile("s_wait_asynccnt 0x0" ::: "memory");
#else
    for (int w = threadIdx.x; w < DW; w += 256) sA[w] = row[w];
    for (int w = DW + threadIdx.x; w < FRW; w += 256) sA[w] = {0.f, 0.f};
#endif
    __syncthreads();
    c32* res = block_fft<2048, false>(sA, sB);
    const size_t base = (size_t)line * FRW;
    for (int w = threadIdx.x; w < FRW; w += 256) {
        float minus = nnl_out[base + w] - mul_out[base + w];
        out[base + w] = { minus * kC1 + res[w].x * kC2, res[w].y * kC2 };
    }
}

// ---------------------------------------------------------------------------
extern "C" void kernel_launch(void* const* d_in, const int* in_sizes, int n_in,
                              void* d_out, int out_size, void* d_ws, size_t ws_size,
                              hipStream_t stream)
{
    const float* nnline_output = (const float*)d_in[0];   // [B,2048,2048]
    const float* multi_output  = (const float*)d_in[1];   // [B,2048,2048]
    const float* input         = (const float*)d_in[2];   // [B,1024,1024]
    const float* nnline_input  = (const float*)d_in[3];   // [B,1024,1024]

    char* ws   = (char*)d_ws;
    c32*  spc  = (c32*)(ws);                                        // 32 MiB
    c32*  escl = (c32*)(ws + (size_t)32 * 1024 * 1024);             // 32 KiB
    c32*  rec1 = (c32*)(ws + (size_t)32 * 1024 * 1024 + 65536);     // 64 MiB
    c32*  out  = (c32*)d_out;                                       // complex64

    k1_row_ifft1024     <<<dim3(BATCH * DH),  dim3(256), 0, stream>>>(nnline_input, spc);
    k2_col_fft_phi      <<<dim3(BATCH * DW),  dim3(256), 0, stream>>>(spc, input, escl);
    k3_col_ifft2048     <<<dim3(BATCH * DW),  dim3(256), 0, stream>>>(input, escl, rec1);
    k4_row_fft2048_combine<<<dim3(BATCH * FRH), dim3(256), 0, stream>>>(rec1, nnline_output,
                                                                        multi_output, out);
}